// Net_34248069218573
// MI455X (gfx1250) — compile-verified
//
#include <hip/hip_runtime.h>

// CDNA5 / gfx1250 wave32 implementation of the NeuralCDM-style forward pass.
// GEMMs use V_WMMA_F32_16X16X4_F32 with 4-way N register blocking (16x64 C
// strip per wave). Leading dimensions are template parameters so all fragment
// loads become immediate-offset addressing off two base pointers -> the
// scheduler can clause loads and stagger s_wait_loadcnt instead of draining
// LOADcnt before every WMMA. The sigmoid-inside-reduction stage is VALU
// (v_exp_f32) since it cannot be factored into a matmul.

typedef float v2f __attribute__((ext_vector_type(2)));
typedef float v8f __attribute__((ext_vector_type(8)));

#define BN 256   // batch
#define KN 512   // knowledge_n
#define DN 64    // low_dim

__device__ __forceinline__ float sigmoidf(float x) {
    return 1.0f / (1.0f + __expf(-x));
}

// WMMA f32 GEMM:  C[m, n] = sum_kk A[row(m), aoff+kk] * W[n, woff+kk]
//   row(m) = idx ? idx[m] : m        (gather support for embedding lookup)
// Each wave computes a 16x64 strip of C (4 adjacent 16x16 tiles), K advanced
// 4 per v_wmma_f32_16x16x4_f32.  Fragment layouts per CDNA5 ISA 7.12.2:
//   A 16x4 : lanes 0-15 rows M, lane-half selects K pair {0,1}/{2,3}
//   B 4x16 : lanes hold column N=lane&15, lane-half selects K pair
//   C 16x16: VGPR v -> row v + 8*half, col = lane&15
template<int LDA, int LDW>
__global__ void __launch_bounds__(256)
wmma_gemm_f32(const float* __restrict__ A, int aoff,
              const int* __restrict__ idx,
              const float* __restrict__ W, int woff,
              float* __restrict__ C,
              int tilesN64, int KK, int applySig)
{
    const int wave = (blockIdx.x * blockDim.x + threadIdx.x) >> 5;
    const int lane = threadIdx.x & 31;
    const int tM   = wave / tilesN64;       // 16-row strip index
    const int tN   = wave % tilesN64;       // 64-col strip index
    const int half = lane >> 4;             // 0: K={0,1}, 1: K={2,3} per step
    const int mr   = lane & 15;

    int arow = tM * 16 + mr;
    if (idx) arow = idx[arow];              // wave-uniform branch, lane gather

    const float* aptr = A + (size_t)arow * LDA + aoff + 2 * half;
    // Single B base; the 16/32/48-row tile offsets are compile-time constants
    // (16*LDW etc.) and fold into 24-bit instruction immediate offsets.
    const float* wptr = W + (size_t)(tN * 64 + mr) * LDW + woff + 2 * half;

    v8f acc0 = {0.f,0.f,0.f,0.f,0.f,0.f,0.f,0.f};
    v8f acc1 = acc0, acc2 = acc0, acc3 = acc0;

#pragma unroll 4
    for (int kk = 0; kk < KK; kk += 4) {
        v2f a  = *(const v2f*)(aptr + kk);
        v2f b0 = *(const v2f*)(wptr + kk);
        v2f b1 = *(const v2f*)(wptr + 16 * LDW + kk);
        v2f b2 = *(const v2f*)(wptr + 32 * LDW + kk);
        v2f b3v = *(const v2f*)(wptr + 48 * LDW + kk);
        // 8 args: (neg_a, A, neg_b, B, c_mod, C, reuse_a, reuse_b)
        acc0 = __builtin_amdgcn_wmma_f32_16x16x4_f32(false, a, false, b0,  (short)0, acc0, false, false);
        acc1 = __builtin_amdgcn_wmma_f32_16x16x4_f32(false, a, false, b1,  (short)0, acc1, false, false);
        acc2 = __builtin_amdgcn_wmma_f32_16x16x4_f32(false, a, false, b2,  (short)0, acc2, false, false);
        acc3 = __builtin_amdgcn_wmma_f32_16x16x4_f32(false, a, false, b3v, (short)0, acc3, false, false);
    }

    float* cp = C + (size_t)(tM * 16 + 8 * half) * KN + tN * 64 + mr;
#pragma unroll
    for (int v = 0; v < 8; ++v) {
        float x0 = acc0[v], x1 = acc1[v], x2 = acc2[v], x3 = acc3[v];
        if (applySig) {
            x0 = sigmoidf(x0); x1 = sigmoidf(x1);
            x2 = sigmoidf(x2); x3 = sigmoidf(x3);
        }
        cp[(size_t)v * KN +  0] = x0;
        cp[(size_t)v * KN + 16] = x1;
        cp[(size_t)v * KN + 32] = x2;
        cp[(size_t)v * KN + 48] = x3;
    }
}

// Final stage: o[b,k] = sigmoid( sum_n (sig(SW1[b,n]+KnW1T[n,k])
//                                     - sig(EW2[b,n]+KnW2T[n,k])) * W3[n] + b3 )
// out[b] = sum_k o*kn_emb / sum_k kn_emb.   One block per b, one thread per k.
__global__ void __launch_bounds__(512)
cdm_final(const float* __restrict__ SW1, const float* __restrict__ EW2,
          const float* __restrict__ KnW1T, const float* __restrict__ KnW2T,
          const float* __restrict__ W3, const float* __restrict__ b3,
          const float* __restrict__ kn_emb, float* __restrict__ out)
{
    __shared__ float sA[KN];
    __shared__ float sB[KN];
    __shared__ float sW[KN];
    __shared__ float red[KN];
    __shared__ float sNum;

    const int b = blockIdx.x;
    const int k = threadIdx.x;

    sA[k] = SW1[(size_t)b * KN + k];
    sB[k] = EW2[(size_t)b * KN + k];
    sW[k] = W3[k];
    __syncthreads();

    float acc = 0.f;
    for (int n = 0; n < KN; ++n) {
        float p = sigmoidf(sA[n] + KnW1T[(size_t)n * KN + k]);  // coalesced over k
        float d = sigmoidf(sB[n] + KnW2T[(size_t)n * KN + k]);
        acc = fmaf(p - d, sW[n], acc);
    }
    float o  = sigmoidf(acc + b3[0]);
    float ke = kn_emb[(size_t)b * KN + k];

    red[k] = o * ke;
    __syncthreads();
    for (int s = KN / 2; s > 0; s >>= 1) {
        if (k < s) red[k] += red[k + s];
        __syncthreads();
    }
    if (k == 0) sNum = red[0];
    __syncthreads();

    red[k] = ke;
    __syncthreads();
    for (int s = KN / 2; s > 0; s >>= 1) {
        if (k < s) red[k] += red[k + s];
        __syncthreads();
    }
    if (k == 0) out[b] = sNum / red[0];
}

extern "C" void kernel_launch(void* const* d_in, const int* in_sizes, int n_in,
                              void* d_out, int out_size, void* d_ws, size_t ws_size,
                              hipStream_t stream)
{
    const int*   stu_id     = (const int*)d_in[0];
    const int*   exer_id    = (const int*)d_in[1];
    const float* kn_emb     = (const float*)d_in[2];
    const float* stu_table  = (const float*)d_in[3];
    const float* exer_table = (const float*)d_in[4];
    const float* kn_table   = (const float*)d_in[5];
    const float* W1         = (const float*)d_in[6];
    const float* W2         = (const float*)d_in[7];
    const float* W3         = (const float*)d_in[8];
    const float* b3         = (const float*)d_in[9];
    float* out = (float*)d_out;

    // workspace layout (bytes): P,Q (256x512), KnW1T,KnW2T (512x512), SW1,EW2 (256x512)
    char* ws = (char*)d_ws;
    float* P     = (float*)(ws);
    float* Q     = (float*)(ws + (size_t)1 * BN * KN * 4);
    float* KnW1T = (float*)(ws + (size_t)2 * BN * KN * 4);
    float* KnW2T = (float*)(ws + (size_t)2 * BN * KN * 4 + (size_t)1 * KN * KN * 4);
    float* SW1   = (float*)(ws + (size_t)2 * BN * KN * 4 + (size_t)2 * KN * KN * 4);
    float* EW2   = (float*)(ws + (size_t)3 * BN * KN * 4 + (size_t)2 * KN * KN * 4);

    const int WPB = 8;                 // waves per block
    const int THREADS = WPB * 32;      // 256 threads = 8 wave32

    // Stage 1: P = sigmoid(stu_table[stu_id] @ kn^T), Q likewise.  M=256,N=512,KK=64
    {
        int blocks = ((BN / 16) * (KN / 64)) / WPB;   // 128 strips / 8 = 16
        wmma_gemm_f32<DN, DN><<<blocks, THREADS, 0, stream>>>(
            stu_table, 0, stu_id, kn_table, 0, P, KN / 64, DN, 1);
        wmma_gemm_f32<DN, DN><<<blocks, THREADS, 0, stream>>>(
            exer_table, 0, exer_id, kn_table, 0, Q, KN / 64, DN, 1);
    }
    // Stage 2: KnW1T[n,k] = sum_d W1[n, K+d] * kn[k,d] (transposed output).  M=N=512,KK=64
    {
        int blocks = ((KN / 16) * (KN / 64)) / WPB;   // 256 strips / 8 = 32
        wmma_gemm_f32<KN + DN, DN><<<blocks, THREADS, 0, stream>>>(
            W1, KN, nullptr, kn_table, 0, KnW1T, KN / 64, DN, 0);
        wmma_gemm_f32<KN + DN, DN><<<blocks, THREADS, 0, stream>>>(
            W2, KN, nullptr, kn_table, 0, KnW2T, KN / 64, DN, 0);
    }
    // Stage 3: SW1 = P @ W1[:,:K]^T, EW2 = Q @ W2[:,:K]^T.  M=256,N=512,KK=512
    {
        int blocks = ((BN / 16) * (KN / 64)) / WPB;   // 16
        wmma_gemm_f32<KN, KN + DN><<<blocks, THREADS, 0, stream>>>(
            P, 0, nullptr, W1, 0, SW1, KN / 64, KN, 0);
        wmma_gemm_f32<KN, KN + DN><<<blocks, THREADS, 0, stream>>>(
            Q, 0, nullptr, W2, 0, EW2, KN / 64, KN, 0);
    }
    // Stage 4: fused sigmoid-difference reduction + masked mean.
    cdm_final<<<BN, KN, 0, stream>>>(SW1, EW2, KnW1T, KnW2T, W3, b3, kn_emb, out);
}